// CostVolumeConcat_88862873354648
// MI455X (gfx1250) — compile-verified
//
#include <hip/hip_runtime.h>

typedef float v4f __attribute__((ext_vector_type(4)));
typedef int   v4i __attribute__((ext_vector_type(4)));
typedef __attribute__((address_space(1))) v4i* gv4i_ptr;   // device-global int4*
typedef __attribute__((address_space(3))) v4i* lv4i_ptr;   // LDS int4*

#define Cc 16      // input channels per tensor
#define Hh 128
#define Ww 512
#define Dd 96      // disparities
#define DPB 8      // disparities per block
#define NTHREADS 256

#if defined(__gfx1250__) && __has_builtin(__builtin_amdgcn_global_load_async_to_lds_b128)
#define HAVE_ASYNC 1
#else
#define HAVE_ASYNC 0
#endif

__global__ __launch_bounds__(NTHREADS) void cost_volume_concat_kernel(
    const float* __restrict__ un_l, const float* __restrict__ un_r,
    const int* __restrict__ dirp, float* __restrict__ out)
{
    // 32 rows x 512 floats = 64 KB: rows 0..15 = un_l[c,h,:], rows 16..31 = un_r[c-16,h,:]
    __shared__ float lds[32 * Ww];

    const int h   = blockIdx.x;      // 0..127
    const int dg  = blockIdx.y;      // 0..11
    const int t   = threadIdx.x;     // 0..255
    const int dir = *dirp;           // +1 or -1 (device scalar)

    // ---- Stage all 32 rows for this h into LDS (16 x b128 per thread) ----
    #pragma unroll
    for (int i = 0; i < 16; ++i) {
        const int f4  = t + NTHREADS * i;    // float4 index 0..4095
        const int row = f4 >> 7;             // 128 float4 per 512-float row
        const int col = (f4 & 127) << 2;     // float offset within row
        const float* src = (row < Cc)
            ? (un_l + ((size_t)(row        * Hh + h)) * Ww + col)
            : (un_r + ((size_t)((row - Cc) * Hh + h)) * Ww + col);
#if HAVE_ASYNC
        // CDNA5 async bulk-copy path: GLOBAL_LOAD_ASYNC_TO_LDS_B128 (ASYNCcnt)
        __builtin_amdgcn_global_load_async_to_lds_b128(
            (gv4i_ptr)(v4i*)src,
            (lv4i_ptr)(v4i*)&lds[f4 * 4],
            0, 0);
#else
        *(v4f*)&lds[f4 * 4] = *(const v4f*)src;
#endif
    }
#if HAVE_ASYNC
  #if __has_builtin(__builtin_amdgcn_s_wait_asynccnt)
    __builtin_amdgcn_s_wait_asynccnt(0);
  #else
    asm volatile("s_wait_asynccnt 0" ::: "memory");
  #endif
#endif
    __syncthreads();

    // ---- Emit DPB disparities: 32 ch x 512 w each, NT b128 stores ----
    #pragma unroll
    for (int dd = 0; dd < DPB; ++dd) {
        const int d = dg * DPB + dd;
        #pragma unroll 4
        for (int i = 0; i < 16; ++i) {
            const int f4 = t + NTHREADS * i;   // over 32*128 float4s
            const int c  = f4 >> 7;            // output channel 0..31
            const int w0 = (f4 & 127) << 2;    // w of first lane element
            v4f v;
            if (c < Cc) {
                // l-part: disparity-independent copy (aligned ds_load_b128)
                v = *(const v4f*)&lds[c * Ww + w0];
            } else {
                // r-part: rotate row by d*dir (per-dword, conflict-free)
                const int base = c * Ww;
                const int s = w0 - d * dir;
                v.x = lds[base + ((s + 0) & (Ww - 1))];
                v.y = lds[base + ((s + 1) & (Ww - 1))];
                v.z = lds[base + ((s + 2) & (Ww - 1))];
                v.w = lds[base + ((s + 3) & (Ww - 1))];
            }
            const size_t off = (((size_t)c * Dd + d) * Hh + h) * (size_t)Ww + w0;
            __builtin_nontemporal_store(v, (v4f*)(out + off));
        }
    }
}

extern "C" void kernel_launch(void* const* d_in, const int* in_sizes, int n_in,
                              void* d_out, int out_size, void* d_ws, size_t ws_size,
                              hipStream_t stream) {
    const float* un_l = (const float*)d_in[0];
    const float* un_r = (const float*)d_in[1];
    const int*   dirp = (const int*)d_in[2];
    float* out = (float*)d_out;

    dim3 grid(Hh, Dd / DPB);   // (128, 12)
    cost_volume_concat_kernel<<<grid, NTHREADS, 0, stream>>>(un_l, un_r, dirp, out);
}